// QueryAndGroup_47064251629654
// MI455X (gfx1250) — compile-verified
//
#include <hip/hip_runtime.h>

// Problem constants (from reference setup_inputs)
#define BB       4
#define NN       16384
#define MM       2048
#define CF       64          // feature channels
#define CO       67          // 3 (xyz diff) + 64 (features)
#define NS       32          // nsample
#define RADIUS2  0.01f       // 0.1^2
#define CHUNK    64          // points per scan iteration = 4 WMMAs

typedef __attribute__((ext_vector_type(2))) float v2f;
typedef __attribute__((ext_vector_type(8))) float v8f;

// One wave (32 lanes) handles a tile of 16 query points:
//  Phase 1: scan all N points in chunks of 64 using 4x V_WMMA_F32_16X16X4_F32
//           (d^2 = |q|^2 - 2 q.p + |p|^2 as a rank-4 GEMM); ballots build a
//           64-bit per-row hit mask popped in scan order into LDS.
//           Uniform early exit once every row has NS hits.
//  Phase 2: gather 3 xyz-diff channels + 64 feature channels; lane = sample
//           index k so every channel store is a fully coalesced 128B line.
__launch_bounds__(32)
__global__ void ballquery_group_fused(const float* __restrict__ xyz,
                                      const float* __restrict__ new_xyz,
                                      const float* __restrict__ feats,
                                      float* __restrict__ out) {
  __shared__ int   idxLds[16 * NS];
  __shared__ float qLds[16][4];     // x, y, z, |q|^2

  const int  lane = threadIdx.x;                // 0..31
  const bool lo   = lane < 16;                  // lane-half predicate (cndmask)
  const int  tile = blockIdx.x;                 // 0 .. B*M/16-1
  const int  b    = tile / (MM / 16);
  const int  m0   = (tile % (MM / 16)) * 16;
  const int  mrow = lane & 15;

  // ---- load query tile -------------------------------------------------
  const float* qp = new_xyz + ((size_t)b * MM + m0 + mrow) * 3;
  const float qx = qp[0], qy = qp[1], qz = qp[2];
  if (lo) {
    qLds[mrow][0] = qx; qLds[mrow][1] = qy; qLds[mrow][2] = qz;
    qLds[mrow][3] = qx * qx + qy * qy + qz * qz;
  }
  __syncthreads();

  // A-matrix 16x4 f32 (2 VGPRs), built branchlessly:
  // lanes 0-15: (K0=qx, K1=qy) for M=lane; lanes 16-31: (K2=qz, K3=1.0).
  v2f amat;
  amat.x = lo ? qx : qz;
  amat.y = lo ? qy : 1.0f;

  // C-matrix: c[r] = |q|^2 of row (r + 8*laneHi), per C/D VGPR layout.
  const int mbase = lo ? 0 : 8;
  v8f cacc;
#pragma unroll
  for (int r = 0; r < 8; ++r) cacc[r] = qLds[mbase + r][3];

  const int   row = mrow;
  int         cnt = lo ? 0 : NS;                // upper lanes report "done"
  const float* xb = xyz + (size_t)b * NN * 3;

  // ---- phase 1: WMMA distance scan, 64 points / iteration --------------
  for (int n0 = 0; n0 < NN; n0 += CHUNK) {
    // Batch the four 16-point column loads up front (latency overlap).
    float px[4], py[4], pz[4];
#pragma unroll
    for (int j = 0; j < 4; ++j) {
      const float* pp = xb + (size_t)(n0 + 16 * j + mrow) * 3;
      px[j] = pp[0]; py[j] = pp[1]; pz[j] = pp[2];
    }
    if (n0 + CHUNK < NN) {
      __builtin_prefetch(xb + (size_t)(n0 + CHUNK + lane) * 3, 0, 1);
      __builtin_prefetch(xb + (size_t)(n0 + CHUNK + 32 + lane) * 3, 0, 1);
    }

    // 4 back-to-back rank-4 WMMAs (shared A and C; no D->A/B hazards).
    unsigned mk[4][8];                           // wave-uniform hit masks
#pragma unroll
    for (int j = 0; j < 4; ++j) {
      // B-matrix 4x16 f32, branchless (v_cndmask, no exec manipulation):
      // lanes 0-15 (N=lane): (K0=-2px, K1=-2py);
      // lanes 16-31 (N=lane-16): (K2=-2pz, K3=|p|^2).
      const float pn = px[j] * px[j] + py[j] * py[j] + pz[j] * pz[j];
      v2f bmat;
      bmat.x = -2.0f * (lo ? px[j] : pz[j]);
      bmat.y = lo ? (-2.0f * py[j]) : pn;
      v8f d = __builtin_amdgcn_wmma_f32_16x16x4_f32(
          false, amat, false, bmat, (short)0, cacc, false, false);
      // mk[j][r]: bits[15:0] -> row r, bits[31:16] -> row r+8 (cols 16j..16j+15)
#pragma unroll
      for (int r = 0; r < 8; ++r)
        mk[j][r] = __builtin_amdgcn_ballot_w32(d[r] < RADIUS2);
    }

    unsigned any = 0;
#pragma unroll
    for (int j = 0; j < 4; ++j)
#pragma unroll
      for (int r = 0; r < 8; ++r) any |= mk[j][r];

    if (any) {                                   // uniform branch
      if (lo) {
        // Assemble this row's 64-bit hit mask (bit i -> point n0+i, in order)
        const unsigned r3 = (unsigned)(row & 7);
        unsigned long long m64 = 0ull;
#pragma unroll
        for (int j = 0; j < 4; ++j) {
          unsigned sel = mk[j][0];
#pragma unroll
          for (int r = 1; r < 8; ++r) sel = (r3 == (unsigned)r) ? mk[j][r] : sel;
          const unsigned h16 = (row < 8) ? (sel & 0xFFFFu) : (sel >> 16);
          m64 |= (unsigned long long)h16 << (16 * j);
        }
        while (m64 && cnt < NS) {
          const int bit = __builtin_ctzll(m64);
          idxLds[row * NS + cnt] = n0 + bit;
          ++cnt;
          m64 &= m64 - 1;
        }
      }
    }
    // uniform early exit once every row has NS hits
    if (__builtin_amdgcn_ballot_w32(cnt >= NS) == 0xFFFFFFFFu) break;
  }

  // ---- pad per reference semantics ------------------------------------
  if (lo) {
    const int first = (cnt > 0) ? idxLds[row * NS] : 0;
    for (int k = cnt; k < NS; ++k) idxLds[row * NS + k] = first;
  }
  __syncthreads();

  // ---- phase 2: gather (lane = sample k; stores fully coalesced) ------
  const int k = lane;                            // NS == warpSize == 32
  const size_t chStride = (size_t)MM * NS;       // per-channel stride in out
  for (int ml = 0; ml < 16; ++ml) {
    const int m  = m0 + ml;
    const int id = idxLds[ml * NS + k];
    const float* pr = xb + (size_t)id * 3;
    const float gx = pr[0] - qLds[ml][0];
    const float gy = pr[1] - qLds[ml][1];
    const float gz = pr[2] - qLds[ml][2];
    float* o = out + (((size_t)b * CO) * MM + m) * NS + k;
    o[0]            = gx;
    o[chStride]     = gy;
    o[2 * chStride] = gz;
    const float* fb = feats + ((size_t)b * CF) * NN + id;
    float* oc = out + (((size_t)b * CO + 3) * MM + m) * NS + k;
#pragma unroll 4
    for (int c = 0; c < CF; ++c)
      oc[(size_t)c * chStride] = fb[(size_t)c * NN];
  }
}

extern "C" void kernel_launch(void* const* d_in, const int* in_sizes, int n_in,
                              void* d_out, int out_size, void* d_ws, size_t ws_size,
                              hipStream_t stream) {
  (void)in_sizes; (void)n_in; (void)out_size; (void)d_ws; (void)ws_size;
  const float* xyz     = (const float*)d_in[0];
  const float* new_xyz = (const float*)d_in[1];
  const float* feats   = (const float*)d_in[2];
  float* out           = (float*)d_out;

  const int tiles = (BB * MM) / 16;              // 512 single-wave workgroups
  ballquery_group_fused<<<dim3(tiles), dim3(32), 0, stream>>>(
      xyz, new_xyz, feats, out);
}